// SimpleGNN_53850299957912
// MI455X (gfx1250) — compile-verified
//
#include <hip/hip_runtime.h>
#include <hip/hip_bf16.h>

#define NN 100000
#define EE 1600000
#define DD 128

typedef __attribute__((ext_vector_type(2))) float v2f;
typedef __attribute__((ext_vector_type(8))) float v8f;

// ---------------------------------------------------------------------------
// Zero a float buffer (workspace is poisoned 0xAA; we must re-zero every call)
// ---------------------------------------------------------------------------
__global__ __launch_bounds__(256)
void gnn_zero_f32(float* __restrict__ p, int n) {
    int i = blockIdx.x * blockDim.x + threadIdx.x;
    if (i < n) p[i] = 0.0f;
}

// ---------------------------------------------------------------------------
// Scatter-add: agg[dst] += x[src]   (E*32 work items, one float4 chunk each)
// Degree counted once per edge (chunk 0) when addDeg != 0.
// agg is 51 MB -> L2-resident on MI455X (192 MB L2) so the f32 atomics stay
// in the L2 atomic units.
// ---------------------------------------------------------------------------
__global__ __launch_bounds__(256)
void gnn_scatter_add(const float* __restrict__ x,
                     const int* __restrict__ src,
                     const int* __restrict__ dst,
                     float* __restrict__ agg,
                     float* __restrict__ deg,
                     int addDeg) {
    unsigned int gid = blockIdx.x * blockDim.x + threadIdx.x;
    if (gid >= (unsigned int)EE * 32u) return;
    int e = gid >> 5;              // edge id
    int c = (gid & 31) * 4;        // feature chunk base (0..124)
    int s = src[e];
    int d = dst[e];
    const float4 v = *(const float4*)(x + (size_t)s * DD + c);
    float* a = agg + (size_t)d * DD + c;
    atomicAdd(a + 0, v.x);
    atomicAdd(a + 1, v.y);
    atomicAdd(a + 2, v.z);
    atomicAdd(a + 3, v.w);
    if (addDeg && c == 0) atomicAdd(deg + d, 1.0f);
}

// ---------------------------------------------------------------------------
// Fused SAGE linear:  out = (agg/deg) @ Wl^T + b + xin @ Wr^T  [+ ReLU]
// Block = 256 threads = 8 waves. Block tile: 16 rows x 128 cols.
// Wave w computes the 16x16 output tile at columns [16w, 16w+16).
// K-loop in steps of 4 using V_WMMA_F32_16X16X4_F32 (fp32 in, fp32 acc).
//
// A-frag layout (ISA 7.12.2, 32-bit A 16x4): lanes 0-15 hold K={k,k+1} of
// row M=lane; lanes 16-31 hold K={k+2,k+3} of row M=lane-16.
// B-frag layout (row of B striped across lanes per VGPR): VGPR0 = row K=k
// (lanes 0-15, N=lane) / row K=k+2 (lanes 16-31); VGPR1 = rows k+1 / k+3.
// B[k][n] = W[n][k] (because out = in @ W^T).
// ---------------------------------------------------------------------------
__global__ __launch_bounds__(256)
void gnn_sage_gemm(const float* __restrict__ agg,
                   const float* __restrict__ deg,
                   const float* __restrict__ xin,
                   const float* __restrict__ Wl,
                   const float* __restrict__ bias,
                   const float* __restrict__ Wr,
                   float* __restrict__ out,
                   int doRelu) {
    __shared__ float sA[16][132];   // degree-normalized agg rows (pad: 132%64
    __shared__ float sX[16][132];   // gives conflict-free column reads)

    const int m0  = blockIdx.x * 16;
    const int tid = threadIdx.x;

    // ---- stage 16 rows of agg (scaled by 1/max(deg,1)) and x into LDS ----
    {
        int row = tid >> 4;            // 0..15
        int cb  = (tid & 15) * 8;      // 8 consecutive floats per thread
        float r = 1.0f / fmaxf(deg[m0 + row], 1.0f);
        const float* ag = agg + (size_t)(m0 + row) * DD + cb;
        const float* xr = xin + (size_t)(m0 + row) * DD + cb;
#pragma unroll
        for (int j = 0; j < 8; ++j) {
            sA[row][cb + j] = ag[j] * r;
            sX[row][cb + j] = xr[j];
        }
    }
    __syncthreads();

    const int wave = tid >> 5;
    const int lane = tid & 31;
    const int half = lane >> 4;     // 0: lanes 0-15, 1: lanes 16-31
    const int l15  = lane & 15;
    const int n0   = wave * 16;     // this wave's column tile

    // accumulator starts at the bias (bias depends only on column)
    v8f acc;
    {
        float bj = bias[n0 + l15];
#pragma unroll
        for (int v = 0; v < 8; ++v) acc[v] = bj;
    }

    const float* wlrow = Wl + (size_t)(n0 + l15) * DD;   // W[n][*]
    const float* wrrow = Wr + (size_t)(n0 + l15) * DD;

#pragma unroll 4
    for (int k = 0; k < DD; k += 4) {
        const int ka = k + 2 * half;          // this lane's K pair
        v2f a, ax, bl, br;
        a.x  = sA[l15][ka];     a.y  = sA[l15][ka + 1];
        ax.x = sX[l15][ka];     ax.y = sX[l15][ka + 1];
        bl.x = wlrow[ka];       bl.y = wlrow[ka + 1];
        br.x = wrrow[ka];       br.y = wrrow[ka + 1];
        // acc += A_agg x B_Wl ; acc += A_x x B_Wr
        acc = __builtin_amdgcn_wmma_f32_16x16x4_f32(
                  false, a,  false, bl, (short)0, acc, false, false);
        acc = __builtin_amdgcn_wmma_f32_16x16x4_f32(
                  false, ax, false, br, (short)0, acc, false, false);
    }

    // ---- store: C layout (ISA 7.12.2): VGPR v, lanes 0-15 -> M=v,
    // lanes 16-31 -> M=v+8, N = lane&15 ----
    float* obase = out + (size_t)m0 * DD + n0 + l15;
#pragma unroll
    for (int v = 0; v < 8; ++v) {
        float val = acc[v];
        if (doRelu) val = fmaxf(val, 0.0f);
        obase[(size_t)(v + 8 * half) * DD] = val;
    }
}

// ---------------------------------------------------------------------------
// Orchestration
// ---------------------------------------------------------------------------
extern "C" void kernel_launch(void* const* d_in, const int* in_sizes, int n_in,
                              void* d_out, int out_size, void* d_ws, size_t ws_size,
                              hipStream_t stream) {
    const float* x   = (const float*)d_in[0];
    const int*   ei  = (const int*)d_in[1];     // [2, E]: row0=src, row1=dst
    const float* W1l = (const float*)d_in[2];
    const float* b1  = (const float*)d_in[3];
    const float* W1r = (const float*)d_in[4];
    const float* W2l = (const float*)d_in[5];
    const float* b2  = (const float*)d_in[6];
    const float* W2r = (const float*)d_in[7];
    float* out = (float*)d_out;

    const int* src = ei;
    const int* dst = ei + EE;

    // workspace layout: [ agg (N*D) | deg (N) | h (N*D) ]
    float* agg = (float*)d_ws;
    float* deg = agg + (size_t)NN * DD;
    float* h   = deg + NN;

    const int zThreads = 256;
    const int aggN  = NN * DD;             // 12.8M floats
    const int zAll  = aggN + NN;           // agg + deg contiguous

    const unsigned int scatterItems  = (unsigned int)EE * 32u;
    const unsigned int scatterBlocks = (scatterItems + 255u) / 256u;
    const int gemmBlocks = NN / 16;        // 100000/16 = 6250, exact

    // ---- Layer 1 ----
    gnn_zero_f32<<<(zAll + zThreads - 1) / zThreads, zThreads, 0, stream>>>(agg, zAll);
    gnn_scatter_add<<<scatterBlocks, 256, 0, stream>>>(x, src, dst, agg, deg, 1);
    gnn_sage_gemm<<<gemmBlocks, 256, 0, stream>>>(agg, deg, x, W1l, b1, W1r, h, 1);

    // ---- Layer 2 (degrees unchanged) ----
    gnn_zero_f32<<<(aggN + zThreads - 1) / zThreads, zThreads, 0, stream>>>(agg, aggN);
    gnn_scatter_add<<<scatterBlocks, 256, 0, stream>>>(h, src, dst, agg, deg, 0);
    gnn_sage_gemm<<<gemmBlocks, 256, 0, stream>>>(agg, deg, h, W2l, b2, W2r, out, 0);
}